// SemiGCNConv_18159121728107
// MI455X (gfx1250) — compile-verified
//
#include <hip/hip_runtime.h>

typedef __attribute__((ext_vector_type(2))) float v2f;
typedef __attribute__((ext_vector_type(8))) float v8f;

#define NNODES 50000
#define NEDGES 800000
#define DIM    64

// ---- degree with self loop: deg[v] = 1 + #edges with dst==v ----
__global__ void k_init_deg(float* __restrict__ deg) {
    int i = blockIdx.x * blockDim.x + threadIdx.x;
    if (i < NNODES) deg[i] = 1.0f;
}

__global__ void k_count_deg(const int* __restrict__ dst, float* __restrict__ deg) {
    int i = blockIdx.x * blockDim.x + threadIdx.x;
    if (i < NEDGES) atomicAdd(&deg[dst[i]], 1.0f);
}

__global__ void k_dinv(const float* __restrict__ deg, float* __restrict__ dinv) {
    int i = blockIdx.x * blockDim.x + threadIdx.x;
    if (i < NNODES) dinv[i] = rsqrtf(deg[i]);   // deg >= 1 always
}

// ---- g = (x @ W) * dinv[row] via FP32 WMMA; also seed acc with self-loop term ----
// One wave computes one 16x16 output tile. 3125 row tiles * 4 col tiles = 12500 waves.
__global__ void k_gemm_scale(const float* __restrict__ x, const float* __restrict__ W,
                             const float* __restrict__ dinv,
                             float* __restrict__ g, float* __restrict__ acc) {
    const int NT = (NNODES / 16) * (DIM / 16);          // 12500
    int wave = (blockIdx.x * blockDim.x + threadIdx.x) >> 5;
    int lane = threadIdx.x & 31;
    if (wave >= NT) return;                             // wave-uniform: EXEC stays all-ones

    int rt = wave >> 2;                                 // row tile (0..3124)
    int ct = wave & 3;                                  // col tile (0..3)
    int m0 = rt * 16, n0 = ct * 16;
    int half = lane >> 4;                               // 0: K={0,1}, 1: K={2,3} within step
    int l    = lane & 15;

    const float* xrow = x + (size_t)(m0 + l) * DIM + half * 2;  // A: row m0+l
    const float* wcol = W + n0 + l;                              // B: column n0+l

    v8f c = {};
#pragma unroll
    for (int k = 0; k < DIM; k += 4) {
        v2f a, b;
        // A-matrix 16x4 f32 layout: lanes 0-15 K=k+{0,1}; lanes 16-31 K=k+{2,3}
        a.x = xrow[k];
        a.y = xrow[k + 1];
        // B-matrix 4x16 f32 layout: VGPR0 -> K=k+2*half, VGPR1 -> K=k+2*half+1, N=lane&15
        int kb = k + half * 2;
        b.x = wcol[(size_t)kb * DIM];
        b.y = wcol[(size_t)(kb + 1) * DIM];
        c = __builtin_amdgcn_wmma_f32_16x16x4_f32(
                /*neg_a=*/false, a, /*neg_b=*/false, b,
                /*c_mod=*/(short)0, c, /*reuse_a=*/false, /*reuse_b=*/false);
    }

    // C/D layout: VGPR v -> M = v + 8*half, N = lane&15
#pragma unroll
    for (int v = 0; v < 8; ++v) {
        int row = m0 + v + half * 8;
        float gv = c[v] * dinv[row];
        size_t idx = (size_t)row * DIM + n0 + l;
        g[idx]   = gv;   // message value (already has dinv[src] folded in)
        acc[idx] = gv;   // self-loop contribution seeds the accumulator
    }
}

// ---- scatter-add: acc[dst] += g[src], 64 channels per edge ----
__global__ void k_scatter(const int* __restrict__ src, const int* __restrict__ dst,
                          const float* __restrict__ g, float* __restrict__ acc) {
    long long tid = (long long)blockIdx.x * blockDim.x + threadIdx.x;
    int e  = (int)(tid >> 6);
    int ch = (int)(tid & 63);
    if (e < NEDGES) {
        int s = src[e], d = dst[e];
        atomicAdd(&acc[(size_t)d * DIM + ch], g[(size_t)s * DIM + ch]);
    }
}

// ---- out = relu(acc * dinv[dst] + bias), in place on d_out ----
__global__ void k_final(float* __restrict__ out, const float* __restrict__ dinv,
                        const float* __restrict__ bias) {
    int tid = blockIdx.x * blockDim.x + threadIdx.x;
    if (tid < NNODES * DIM) {
        int v = tid >> 6, ch = tid & 63;
        float val = out[tid] * dinv[v] + bias[ch];
        out[tid] = fmaxf(val, 0.0f);
    }
}

extern "C" void kernel_launch(void* const* d_in, const int* in_sizes, int n_in,
                              void* d_out, int out_size, void* d_ws, size_t ws_size,
                              hipStream_t stream) {
    const float* x  = (const float*)d_in[0];   // [N, 64]
    const int*   ei = (const int*)d_in[1];     // [2, E] row-major
    const float* W  = (const float*)d_in[2];   // [64, 64]
    const float* b  = (const float*)d_in[3];   // [64]
    const int* src = ei;                       // edge_index[0]
    const int* dst = ei + NEDGES;              // edge_index[1]

    float* out  = (float*)d_out;               // doubles as accumulator
    float* deg  = (float*)d_ws;                // N floats
    float* dinv = deg + NNODES;                // N floats
    float* g    = dinv + NNODES;               // N*64 floats

    k_init_deg <<<(NNODES + 255) / 256, 256, 0, stream>>>(deg);
    k_count_deg<<<(NEDGES + 255) / 256, 256, 0, stream>>>(dst, deg);
    k_dinv     <<<(NNODES + 255) / 256, 256, 0, stream>>>(deg, dinv);

    const int nt = (NNODES / 16) * (DIM / 16);          // 12500 tiles, 1 wave each
    const int wpb = 256 / 32;                           // 8 waves per block
    k_gemm_scale<<<(nt + wpb - 1) / wpb, 256, 0, stream>>>(x, W, dinv, g, out);

    long long sthreads = (long long)NEDGES * DIM;       // 51.2M
    k_scatter<<<(int)((sthreads + 255) / 256), 256, 0, stream>>>(src, dst, g, out);

    k_final<<<(NNODES * DIM + 255) / 256, 256, 0, stream>>>(out, dinv, b);
}